// Subsampling_Layer_8014408974464
// MI455X (gfx1250) — compile-verified
//
#include <hip/hip_runtime.h>
#include <math.h>

// ---------------------------------------------------------------------------
// MRI non-Cartesian forward model (NUFFT-like) on gfx1250, f32 WMMA.
//   out[c,h,w] = normalized | sum_m conj(Eh[m,h]) ksp[c,m] conj(Ew[m,w]) |
//   ksp[c,m]   = sum_h Eh[m,h] * sum_w img[c,h,w] * Ew[m,w]
// ~105 GFLOP fp32 -> ~79 GFLOP on the matrix pipe via Gauss 3-mult complex
// GEMM. Working set (< 50MB) is L2-resident (192MB); HBM traffic negligible
// at 23.3 TB/s => purely compute-bound on V_WMMA_F32_16X16X4_F32.
// Planar (split re/im) tables so fragment loads are direct b64 vector loads
// with zero register repacking.
// ---------------------------------------------------------------------------

typedef float v2f __attribute__((ext_vector_type(2)));
typedef float v8f __attribute__((ext_vector_type(8)));

#define RESN     320
#define NSHOTS   8
#define SPS      1001
#define M_TOTAL  (NSHOTS * SPS)   // 8008 samples
#define M_PAD    8016             // 501 tiles of 16 (pad rows zeroed)
#define NC       8                // coils
#define TWO_PI   6.283185307179586f
#define RB       256              // reduction blocks

#define V8F_ZERO {0.f, 0.f, 0.f, 0.f, 0.f, 0.f, 0.f, 0.f}

__device__ __forceinline__ v8f wmma_f32(v2f a, v2f b, v8f c) {
  // V_WMMA_F32_16X16X4_F32 : D = A(16x4) * B(4x16) + C(16x16)
  return __builtin_amdgcn_wmma_f32_16x16x4_f32(
      /*neg_a=*/false, a, /*neg_b=*/false, b,
      /*c_mod=*/(short)0, c, /*reuse_a=*/false, /*reuse_b=*/false);
}

__device__ __forceinline__ v2f ldv2(const float* __restrict__ p) {
  return *(const v2f*)p;   // callers guarantee 8-byte alignment (even offsets)
}

// ---------------------------------------------------------------------------
// 1) Cubic Hermite resampling of trajectory (uniform knots, gap=10).
// ---------------------------------------------------------------------------
__global__ void traj_kernel(const float* __restrict__ x, float* __restrict__ xt) {
  int gid = blockIdx.x * blockDim.x + threadIdx.x;
  if (gid >= M_TOTAL * 2) return;
  int d    = gid & 1;
  int m    = gid >> 1;
  int shot = m / SPS;
  int tt   = m - shot * SPS;
  int I    = (tt == 0) ? 0 : (tt - 1) / 10;
  if (I > 99) I = 99;
  const float* base = x + (size_t)shot * SPS * 2 + d;
  float y0 = base[(I) * 20];
  float y1 = base[(I + 1) * 20];
  float dI = (y1 - y0) * 0.1f;
  float mI, mI1;
  if (I == 0) { mI = dI; }
  else        { float ym = base[(I - 1) * 20]; mI = 0.5f * (dI + (y0 - ym) * 0.1f); }
  if (I == 99) { mI1 = dI; }
  else         { float y2 = base[(I + 2) * 20]; mI1 = 0.5f * (dI + (y2 - y1) * 0.1f); }
  float s  = (float)(tt - 10 * I) * 0.1f;
  float s2 = s * s, s3 = s2 * s;
  float h00 = 1.0f - 3.0f * s2 + 2.0f * s3;
  float h10 = s - 2.0f * s2 + s3;
  float h01 = 3.0f * s2 - 2.0f * s3;
  float h11 = s3 - s2;
  xt[gid] = h00 * y0 + h10 * mI * 10.0f + h01 * y1 + h11 * mI1 * 10.0f;
}

// ---------------------------------------------------------------------------
// 2) Planar phase tables: Eh[m,n] = exp(-2i*pi*xt_h[m]*(n-160)/320), ditto Ew.
//    Pad rows (m >= 8008) zeroed so the GEMM1 M-tail tile needs no predication
//    (WMMA requires EXEC all-ones).
// ---------------------------------------------------------------------------
__global__ void phase_kernel(const float* __restrict__ xt,
                             float* __restrict__ EhR, float* __restrict__ EhI,
                             float* __restrict__ EwR, float* __restrict__ EwI) {
  int gid = blockIdx.x * blockDim.x + threadIdx.x;
  if (gid >= M_PAD * RESN) return;
  int m = gid / RESN;
  int n = gid - m * RESN;
  float ehr = 0.f, ehi = 0.f, ewr = 0.f, ewi = 0.f;
  if (m < M_TOTAL) {
    float k = (TWO_PI / (float)RESN) * ((float)n - 160.0f);
    float sh, ch, sw, cw;
    __sincosf(xt[2 * m + 0] * k, &sh, &ch);
    __sincosf(xt[2 * m + 1] * k, &sw, &cw);
    ehr = ch; ehi = -sh; ewr = cw; ewi = -sw;
  }
  EhR[gid] = ehr; EhI[gid] = ehi;
  EwR[gid] = ewr; EwI[gid] = ewi;
}

// ---------------------------------------------------------------------------
// 2b) Split interleaved-complex image into planar re/im.
// ---------------------------------------------------------------------------
__global__ void split_img_kernel(const float* __restrict__ image,
                                 float* __restrict__ imgR, float* __restrict__ imgI) {
  int i = blockIdx.x * blockDim.x + threadIdx.x;
  if (i < NC * RESN * RESN) {
    imgR[i] = image[2 * i];
    imgI[i] = image[2 * i + 1];
  }
}

// ---------------------------------------------------------------------------
// 3) GEMM1 + fold: one wave per (16-m tile, coil).
//    T(16m x 16h) = Ew(16m x 320w) * img^T(320w x 16h) via Gauss 3-mult
//    complex WMMA, then ksp[c,m] = sum_h Eh[m,h]*T[m,h] by shfl-xor reduce
//    over the 16-lane N-striping of the C layout. Tables are L2-resident.
// ---------------------------------------------------------------------------
__global__ __launch_bounds__(32) void gemm1_kernel(
    const float* __restrict__ imgR, const float* __restrict__ imgI,
    const float* __restrict__ EhR, const float* __restrict__ EhI,
    const float* __restrict__ EwR, const float* __restrict__ EwI,
    float* __restrict__ kspR, float* __restrict__ kspI) {
  const int m0   = blockIdx.x * 16;
  const int c    = blockIdx.y;
  const int l    = threadIdx.x;
  const int half = l >> 4;     // K sub-block select (A/B fragment layouts)
  const int lm   = l & 15;
  float accR[8], accI[8];
#pragma unroll
  for (int v = 0; v < 8; ++v) { accR[v] = 0.0f; accI[v] = 0.0f; }

  const size_t aRow = (size_t)(m0 + lm) * RESN + 2 * half;  // A: M=m, K=w

  for (int ht = 0; ht < 20; ++ht) {
    const int h0 = ht * 16;
    const size_t bRow = ((size_t)c * RESN + h0 + lm) * RESN + 2 * half; // B: K=w, N=h
    v8f C1 = V8F_ZERO;   // Ar*Br
    v8f C2 = V8F_ZERO;   // Ai*Bi
    v8f C3 = V8F_ZERO;   // (Ar+Ai)*(Br+Bi)
    for (int w0 = 0; w0 < RESN; w0 += 4) {
      v2f Ar = ldv2(EwR + aRow + w0);
      v2f Ai = ldv2(EwI + aRow + w0);
      v2f Br = ldv2(imgR + bRow + w0);
      v2f Bi = ldv2(imgI + bRow + w0);
      v2f As = Ar + Ai;
      v2f Bs = Br + Bi;
      C1 = wmma_f32(Ar, Br, C1);
      C2 = wmma_f32(Ai, Bi, C2);
      C3 = wmma_f32(As, Bs, C3);
    }
    // Fold Eh and reduce over the 16 h-columns held across lanes.
#pragma unroll
    for (int v = 0; v < 8; ++v) {
      const int ml = v + 8 * half;                        // C layout: M = v (+8)
      const size_t ei = (size_t)(m0 + ml) * RESN + h0 + lm; // N = lane%16
      float er = EhR[ei], eim = EhI[ei];
      float tr = C1[v] - C2[v];                           // Gauss recombine
      float ti = C3[v] - C1[v] - C2[v];
      float pr = er * tr - eim * ti;
      float pi = er * ti + eim * tr;
#pragma unroll
      for (int mask = 8; mask >= 1; mask >>= 1) {
        pr += __shfl_xor(pr, mask, 32);
        pi += __shfl_xor(pi, mask, 32);
      }
      accR[v] += pr; accI[v] += pi;
    }
  }
  if (lm == 0) {
#pragma unroll
    for (int v = 0; v < 8; ++v) {
      const int m = m0 + v + 8 * half;
      if (m < M_TOTAL) {
        kspR[(size_t)c * M_PAD + m] = accR[v];
        kspI[(size_t)c * M_PAD + m] = accI[v];
      }
    }
  }
}

// ---------------------------------------------------------------------------
// 4) GEMM2: rec(2560 x 320) = G(2560 x 8008) * conj(Ew)(8008 x 320), complex.
//    G[r,m] = conj(Eh[m,h])*ksp[c,m] built on the fly into planar LDS.
//    Block tile 64x32, 8 waves x (16x16 complex tile), K slab = 32,
//    Gauss 3-mult accumulators; epilogue writes |rec| directly to d_out.
// ---------------------------------------------------------------------------
#define KBLK 32
#define LSTR 34   // LDS row stride (even for b64 align, spreads banks)
__global__ __launch_bounds__(256) void gemm2_kernel(
    const float* __restrict__ EhR, const float* __restrict__ EhI,
    const float* __restrict__ EwR, const float* __restrict__ EwI,
    const float* __restrict__ kspR, const float* __restrict__ kspI,
    float* __restrict__ out) {
  __shared__ float GtR[64 * LSTR];   // [r][k]
  __shared__ float GtI[64 * LSTR];
  __shared__ float BtR[32 * LSTR];   // [n][k]  (K contiguous => b64 frag reads)
  __shared__ float BtI[32 * LSTR];
  const int tid  = threadIdx.x;
  const int wave = tid >> 5;
  const int l    = tid & 31;
  const int half = l >> 4;
  const int lm   = l & 15;
  const int wr   = wave >> 1;           // 4 row sub-tiles
  const int wc   = wave & 1;            // 2 col sub-tiles
  const int rowTile = blockIdx.x;       // 0..39 (64 rows; never crosses coils)
  const int colTile = blockIdx.y;       // 0..9  (32 cols)
  const int c  = rowTile / 5;
  const int h0 = (rowTile % 5) * 64;
  const int w0 = colTile * 32;

  v8f C1 = V8F_ZERO;   // Ar*Br
  v8f C2 = V8F_ZERO;   // Ai*Bi
  v8f C3 = V8F_ZERO;   // (Ar+Ai)*(Br+Bi)

  const int aOff = (wr * 16 + lm) * LSTR + 2 * half;
  const int bOff = (wc * 16 + lm) * LSTR + 2 * half;

  for (int kb = 0; kb < (M_TOTAL + KBLK - 1) / KBLK; ++kb) {
    const int k0 = kb * KBLK;
    // Stage B2 slab: conj(Ew[m, w0+n]) -> Bt[n][k]  (coalesced over n)
    for (int i = tid; i < KBLK * 32; i += 256) {
      const int n = i & 31, k = i >> 5;
      const int m = k0 + k;
      float br = 0.f, bi = 0.f;
      if (m < M_TOTAL) {
        const size_t gi = (size_t)m * RESN + w0 + n;
        br = EwR[gi];
        bi = -EwI[gi];
      }
      BtR[n * LSTR + k] = br;
      BtI[n * LSTR + k] = bi;
    }
    // Stage G slab: conj(Eh[m, h0+r]) * ksp[c, m] -> Gt[r][k] (coalesced over r)
    for (int i = tid; i < 64 * KBLK; i += 256) {
      const int r = i & 63, k = i >> 6;
      const int m = k0 + k;
      float gr = 0.f, gi = 0.f;
      if (m < M_TOTAL) {
        const size_t ei = (size_t)m * RESN + h0 + r;
        float er = EhR[ei], eim = EhI[ei];
        float kr = kspR[(size_t)c * M_PAD + m];
        float ki = kspI[(size_t)c * M_PAD + m];
        gr = er * kr + eim * ki;     // conj(e)*k
        gi = er * ki - eim * kr;
      }
      GtR[r * LSTR + k] = gr;
      GtI[r * LSTR + k] = gi;
    }
    __syncthreads();
#pragma unroll
    for (int kk = 0; kk < KBLK; kk += 4) {
      v2f Ar = ldv2(GtR + aOff + kk);      // A: M=row, K=m (contiguous pair)
      v2f Ai = ldv2(GtI + aOff + kk);
      v2f Br = ldv2(BtR + bOff + kk);      // B: N=w,   K=m (contiguous pair)
      v2f Bi = ldv2(BtI + bOff + kk);
      v2f As = Ar + Ai;
      v2f Bs = Br + Bi;
      C1 = wmma_f32(Ar, Br, C1);
      C2 = wmma_f32(Ai, Bi, C2);
      C3 = wmma_f32(As, Bs, C3);
    }
    __syncthreads();
  }
  // Epilogue: Gauss recombine + magnitude straight to output (row = c*320+h).
#pragma unroll
  for (int v = 0; v < 8; ++v) {
    const int row = rowTile * 64 + wr * 16 + v + 8 * half;
    const int col = w0 + wc * 16 + lm;
    float rr = C1[v] - C2[v];
    float ri = C3[v] - C1[v] - C2[v];
    out[(size_t)row * RESN + col] = sqrtf(rr * rr + ri * ri);
  }
}

// ---------------------------------------------------------------------------
// 5) Global min/max + normalize.
// ---------------------------------------------------------------------------
__global__ void reduce1_kernel(const float* __restrict__ out, float* __restrict__ red, int n) {
  __shared__ float smn[256], smx[256];
  float mn = 3.402823466e38f, mx = -3.402823466e38f;
  for (int i = blockIdx.x * blockDim.x + threadIdx.x; i < n; i += gridDim.x * blockDim.x) {
    float v = out[i];
    mn = fminf(mn, v); mx = fmaxf(mx, v);
  }
  smn[threadIdx.x] = mn; smx[threadIdx.x] = mx;
  __syncthreads();
  for (int s = 128; s > 0; s >>= 1) {
    if ((int)threadIdx.x < s) {
      smn[threadIdx.x] = fminf(smn[threadIdx.x], smn[threadIdx.x + s]);
      smx[threadIdx.x] = fmaxf(smx[threadIdx.x], smx[threadIdx.x + s]);
    }
    __syncthreads();
  }
  if (threadIdx.x == 0) { red[2 + blockIdx.x] = smn[0]; red[2 + RB + blockIdx.x] = smx[0]; }
}

__global__ void reduce2_kernel(float* __restrict__ red) {
  __shared__ float smn[RB], smx[RB];
  const int t = threadIdx.x;
  smn[t] = red[2 + t]; smx[t] = red[2 + RB + t];
  __syncthreads();
  for (int s = RB / 2; s > 0; s >>= 1) {
    if (t < s) { smn[t] = fminf(smn[t], smn[t + s]); smx[t] = fmaxf(smx[t], smx[t + s]); }
    __syncthreads();
  }
  if (t == 0) { red[0] = smn[0]; red[1] = smx[0]; }
}

__global__ void normalize_kernel(float* __restrict__ out, const float* __restrict__ red, int n) {
  int i = blockIdx.x * blockDim.x + threadIdx.x;
  if (i < n) {
    float mn = red[0];
    float inv = 1.0f / (red[1] - mn + 1e-8f);
    out[i] = (out[i] - mn) * inv;
  }
}

// ---------------------------------------------------------------------------
extern "C" void kernel_launch(void* const* d_in, const int* in_sizes, int n_in,
                              void* d_out, int out_size, void* d_ws, size_t ws_size,
                              hipStream_t stream) {
  const float* image = (const float*)d_in[0];  // (1,8,320,320,2) f32
  const float* x     = (const float*)d_in[1];  // (8,1001,2) f32
  float* out = (float*)d_out;                  // (1,8,320,320) f32 = 819200

  // Planar workspace layout (~48 MB; everything rewritten each call).
  const size_t TBL = (size_t)M_PAD * RESN;     // 2,565,120 (even)
  const size_t IMG = (size_t)NC * RESN * RESN; // 819,200  (even)
  const size_t KSP = (size_t)NC * M_PAD;       // 64,128   (even)
  float* p    = (float*)d_ws;
  float* EhR  = p; p += TBL;
  float* EhI  = p; p += TBL;
  float* EwR  = p; p += TBL;
  float* EwI  = p; p += TBL;
  float* imgR = p; p += IMG;
  float* imgI = p; p += IMG;
  float* kspR = p; p += KSP;
  float* kspI = p; p += KSP;
  float* xt   = p; p += (size_t)M_TOTAL * 2;
  float* red  = p;                              // 2 + 2*RB floats

  const int n = NC * RESN * RESN;  // 819200

  traj_kernel<<<(M_TOTAL * 2 + 255) / 256, 256, 0, stream>>>(x, xt);
  phase_kernel<<<(M_PAD * RESN + 255) / 256, 256, 0, stream>>>(xt, EhR, EhI, EwR, EwI);
  split_img_kernel<<<(n + 255) / 256, 256, 0, stream>>>(image, imgR, imgI);

  dim3 g1(M_PAD / 16, NC);   // 501 x 8 single-wave blocks
  gemm1_kernel<<<g1, 32, 0, stream>>>(imgR, imgI, EhR, EhI, EwR, EwI, kspR, kspI);

  dim3 g2((NC * RESN) / 64, RESN / 32);  // 40 x 10 blocks of 256
  gemm2_kernel<<<g2, 256, 0, stream>>>(EhR, EhI, EwR, EwI, kspR, kspI, out);

  reduce1_kernel<<<RB, 256, 0, stream>>>(out, red, n);
  reduce2_kernel<<<1, RB, 0, stream>>>(red);
  normalize_kernel<<<(n + 255) / 256, 256, 0, stream>>>(out, red, n);
}